// GraphAttentionLayer_78151224918801
// MI455X (gfx1250) — compile-verified
//
#include <hip/hip_runtime.h>
#include <math.h>

#define N_NODES 50000
#define N_EDGES 800000
#define IN_F    128
#define OUT_F   128
#define EDGE_F  16
#define HEADS   4
#define HEAD_D  32

typedef float v2f __attribute__((ext_vector_type(2)));
typedef float v4f __attribute__((ext_vector_type(4)));
typedef float v8f __attribute__((ext_vector_type(8)));

// ---------------------------------------------------------------------------
// Zero the per-(node,head) softmax denominator accumulator in workspace.
// ---------------------------------------------------------------------------
__global__ __launch_bounds__(256) void zero_denom(float* __restrict__ denom, int n) {
    int i = blockIdx.x * blockDim.x + threadIdx.x;
    if (i < n) denom[i] = 0.0f;
}

// ---------------------------------------------------------------------------
// QKV GEMM: O = x @ W, f32 WMMA 16x16x4.
// grid = (3125, 2, 3): x-tile of 16 rows, 64-col stripe, matrix select.
// One wave per block; 4 accumulators (16x64 output tile), K-loop of 32 steps.
// A-frag (16x4 f32): lane m = lane&15; VGPR0 = K = k + 2*(lane>>4), VGPR1 = +1.
// B-frag (4x16 f32): lane n = lane&15; same K striping as A.
// C/D (16x16 f32):  VGPR r -> row r (lanes 0-15) / r+8 (lanes 16-31), col lane&15.
// ---------------------------------------------------------------------------
__global__ __launch_bounds__(32) void qkv_gemm(
    const float* __restrict__ x,
    const float* __restrict__ Wq, const float* __restrict__ Wk, const float* __restrict__ Wv,
    float* __restrict__ Q, float* __restrict__ Kmat, float* __restrict__ V)
{
    const float* W;
    float* O;
    if (blockIdx.z == 0)      { W = Wq; O = Q;    }
    else if (blockIdx.z == 1) { W = Wk; O = Kmat; }
    else                      { W = Wv; O = V;    }

    const int lane = threadIdx.x & 31;
    const int hl   = lane >> 4;        // half-wave: 0 or 1
    const int lm   = lane & 15;        // row (A) / col (B,C)
    const int m0   = blockIdx.x * 16;
    const int n0   = blockIdx.y * 64;

    const float* xrow = x + (size_t)(m0 + lm) * IN_F;

    v8f acc0 = {}, acc1 = {}, acc2 = {}, acc3 = {};

    for (int k = 0; k < IN_F; k += 4) {
        const int kk = k + hl * 2;
        v2f a = *(const v2f*)(xrow + kk);                 // x[m][kk], x[m][kk+1]
        const float* w0 = W + (size_t)kk * OUT_F + n0 + lm;
        const float* w1 = w0 + OUT_F;
        v2f b0 = { w0[0],  w1[0]  };
        v2f b1 = { w0[16], w1[16] };
        v2f b2 = { w0[32], w1[32] };
        v2f b3 = { w0[48], w1[48] };
        acc0 = __builtin_amdgcn_wmma_f32_16x16x4_f32(false, a, false, b0, (short)0, acc0, false, false);
        acc1 = __builtin_amdgcn_wmma_f32_16x16x4_f32(false, a, false, b1, (short)0, acc1, false, false);
        acc2 = __builtin_amdgcn_wmma_f32_16x16x4_f32(false, a, false, b2, (short)0, acc2, false, false);
        acc3 = __builtin_amdgcn_wmma_f32_16x16x4_f32(false, a, false, b3, (short)0, acc3, false, false);
    }

    float* orow = O + (size_t)(m0 + 8 * hl) * OUT_F + n0 + lm;
#pragma unroll
    for (int r = 0; r < 8; ++r) {
        orow[(size_t)r * OUT_F +  0] = acc0[r];
        orow[(size_t)r * OUT_F + 16] = acc1[r];
        orow[(size_t)r * OUT_F + 32] = acc2[r];
        orow[(size_t)r * OUT_F + 48] = acc3[r];
    }
}

// ---------------------------------------------------------------------------
// Per-edge: score[h] = (Q[src,h,:]·K[dst,h,:]) / sqrt(32) + edge_attr[e] @ We[:,h]
// then atomicAdd(exp(score)) into denom[dst,h].  One wave per edge; lane l
// covers cols 4l..4l+3 (head = l>>3); 8-lane xor-shuffle reduction per head.
// ---------------------------------------------------------------------------
__global__ __launch_bounds__(256) void edge_scores(
    const long long* __restrict__ ei,     // [2, N_EDGES] int64
    const float* __restrict__ ea,         // [N_EDGES, EDGE_F]
    const float* __restrict__ We,         // [EDGE_F, HEADS]
    const float* __restrict__ Q, const float* __restrict__ K,
    float* __restrict__ denom)            // [N_NODES, HEADS]
{
    const int e = blockIdx.x * (blockDim.x >> 5) + (threadIdx.x >> 5);
    if (e >= N_EDGES) return;
    const int lane = threadIdx.x & 31;

    const long long src = ei[e];
    const long long dst = ei[N_EDGES + e];

    v4f qa = *(const v4f*)(Q + (size_t)src * OUT_F + lane * 4);
    v4f ka = *(const v4f*)(K + (size_t)dst * OUT_F + lane * 4);
    float s = qa.x * ka.x + qa.y * ka.y + qa.z * ka.z + qa.w * ka.w;

    // reduce the 8 lanes of each head
    s += __shfl_xor(s, 1, 32);
    s += __shfl_xor(s, 2, 32);
    s += __shfl_xor(s, 4, 32);

    if ((lane & 7) == 0) {
        const int h = lane >> 3;
        float bias = 0.0f;
        const float* ev = ea + (size_t)e * EDGE_F;
#pragma unroll
        for (int i = 0; i < EDGE_F; ++i) bias += ev[i] * We[i * HEADS + h];
        const float sc = s * 0.17677669529663687f + bias;   // 1/sqrt(32)
        atomicAdd(&denom[(size_t)dst * HEADS + h], expf(sc));
    }
}

// ---------------------------------------------------------------------------
// Finalize: out[n] = LN( V[n,h,:] * denom/(denom+1e-10) + x[n] ) * gamma + beta
// (exact collapse of the reference's weighted scatter of V[dst]).
// One wave per node; lane l covers cols 4l..4l+3.
// ---------------------------------------------------------------------------
__global__ __launch_bounds__(256) void finalize(
    const float* __restrict__ x, const float* __restrict__ V,
    const float* __restrict__ denom,
    const float* __restrict__ gamma, const float* __restrict__ beta,
    float* __restrict__ out)
{
    const int n = blockIdx.x * (blockDim.x >> 5) + (threadIdx.x >> 5);
    if (n >= N_NODES) return;
    const int lane = threadIdx.x & 31;
    const int c = lane * 4;
    const int h = lane >> 3;

    const float d = denom[(size_t)n * HEADS + h];
    const float scale = d / (d + 1e-10f);

    v4f xv = *(const v4f*)(x + (size_t)n * OUT_F + c);
    v4f vv = *(const v4f*)(V + (size_t)n * OUT_F + c);
    v4f hv;
    hv.x = vv.x * scale + xv.x;
    hv.y = vv.y * scale + xv.y;
    hv.z = vv.z * scale + xv.z;
    hv.w = vv.w * scale + xv.w;

    float psum = hv.x + hv.y + hv.z + hv.w;
#pragma unroll
    for (int off = 16; off > 0; off >>= 1) psum += __shfl_xor(psum, off, 32);
    const float mu = psum * (1.0f / OUT_F);

    const float d0 = hv.x - mu, d1 = hv.y - mu, d2 = hv.z - mu, d3 = hv.w - mu;
    float pvar = d0 * d0 + d1 * d1 + d2 * d2 + d3 * d3;
#pragma unroll
    for (int off = 16; off > 0; off >>= 1) pvar += __shfl_xor(pvar, off, 32);
    const float inv = rsqrtf(pvar * (1.0f / OUT_F) + 1e-5f);

    v4f g = *(const v4f*)(gamma + c);
    v4f b = *(const v4f*)(beta + c);
    v4f o;
    o.x = d0 * inv * g.x + b.x;
    o.y = d1 * inv * g.y + b.y;
    o.z = d2 * inv * g.z + b.z;
    o.w = d3 * inv * g.w + b.w;
    *(v4f*)(out + (size_t)n * OUT_F + c) = o;
}

// ---------------------------------------------------------------------------
extern "C" void kernel_launch(void* const* d_in, const int* in_sizes, int n_in,
                              void* d_out, int out_size, void* d_ws, size_t ws_size,
                              hipStream_t stream) {
    const float*     x     = (const float*)d_in[0];
    const long long* ei    = (const long long*)d_in[1];
    const float*     ea    = (const float*)d_in[2];
    const float*     Wq    = (const float*)d_in[3];
    const float*     Wk    = (const float*)d_in[4];
    const float*     Wv    = (const float*)d_in[5];
    const float*     We    = (const float*)d_in[6];
    const float*     gamma = (const float*)d_in[7];
    const float*     beta  = (const float*)d_in[8];
    float*           out   = (float*)d_out;

    // workspace layout: Q | K | V | denom
    const size_t mat = (size_t)N_NODES * OUT_F;        // 6.4M floats each
    float* Q     = (float*)d_ws;
    float* Kmat  = Q + mat;
    float* V     = Kmat + mat;
    float* denom = V + mat;                            // N_NODES * HEADS floats

    const int ndenom = N_NODES * HEADS;
    zero_denom<<<(ndenom + 255) / 256, 256, 0, stream>>>(denom, ndenom);

    dim3 ggrid(N_NODES / 16, OUT_F / 64, 3);           // 3125 x 2 x 3
    qkv_gemm<<<ggrid, 32, 0, stream>>>(x, Wq, Wk, Wv, Q, Kmat, V);

    edge_scores<<<(N_EDGES + 7) / 8, 256, 0, stream>>>(ei, ea, We, Q, Kmat, denom);

    finalize<<<(N_NODES + 7) / 8, 256, 0, stream>>>(x, V, denom, gamma, beta, out);
}